// FlightGNNWithGAT_8392366096477
// MI455X (gfx1250) — compile-verified
//
#include <hip/hip_runtime.h>
#include <hip/hip_bf16.h>

typedef __attribute__((ext_vector_type(2))) float v2f;
typedef __attribute__((ext_vector_type(8))) float v8f;

#define NEG_SLOPE 0.2f

// Monotone order-preserving float<->uint mapping for atomicMax on floats.
__device__ __forceinline__ unsigned enc_f(float f) {
    unsigned u = __float_as_uint(f);
    return (u >> 31) ? ~u : (u | 0x80000000u);
}
__device__ __forceinline__ float dec_f(unsigned u) {
    u = (u >> 31) ? (u & 0x7FFFFFFFu) : ~u;
    return __uint_as_float(u);
}

// ---------------------------------------------------------------------------
// Params block (512 floats in ws):
// [0]=cs, [1]=cd, [2]=mlp_b
// [16..31]=v (lin_edge_W @ att_edge), [32..47]=u (mlp_W[64:80])
// [64..95]=W, [96..127]=bias, [128..159]=m0 (mlp_W[0:32]), [160..191]=m1
// [256..511]=bimg: per-(chunk i, lane) WMMA B-operand image, float2 each:
//            bimg[(i*32+lane)*2 + {0,1}] = B_i[k, laneN] for k = 4i+2*half+{0,1}
//            where column 0 = v, column 1 = u, other columns zero.
// ---------------------------------------------------------------------------
__global__ void prep_kernel(const float* __restrict__ W, const float* __restrict__ att_src,
                            const float* __restrict__ att_dst, const float* __restrict__ linW,
                            const float* __restrict__ attE, const float* __restrict__ bias,
                            const float* __restrict__ mlpW, const float* __restrict__ mlpb,
                            float* __restrict__ params) {
    __shared__ float sv[16], su[16];
    const int t = threadIdx.x;  // blockDim.x == 128
    if (t < 16) {
        float v = 0.f;
        #pragma unroll
        for (int c = 0; c < 32; ++c) v += linW[t * 32 + c] * attE[c];
        sv[t] = v;
        su[t] = mlpW[64 + t];
        params[16 + t] = v;
        params[32 + t] = su[t];
    }
    if (t == 16) {
        float cs = 0.f;
        #pragma unroll
        for (int c = 0; c < 32; ++c) cs += W[c] * att_src[c];
        params[0] = cs;
    }
    if (t == 17) {
        float cd = 0.f;
        #pragma unroll
        for (int c = 0; c < 32; ++c) cd += W[c] * att_dst[c];
        params[1] = cd;
    }
    if (t == 18) params[2] = mlpb[0];
    if (t < 32) {
        params[64 + t]  = W[t];
        params[96 + t]  = bias[t];
        params[128 + t] = mlpW[t];
        params[160 + t] = mlpW[32 + t];
    }
    __syncthreads();
    // t = i*32 + lane for i in [0,4), lane in [0,32)
    {
        const int lane  = t & 31;
        const int i     = t >> 5;
        const int laneN = lane & 15;
        const int half  = lane >> 4;
        const int k     = 4 * i + 2 * half;
        const float b0 = (laneN == 0) ? sv[k]     : (laneN == 1) ? su[k]     : 0.f;
        const float b1 = (laneN == 0) ? sv[k + 1] : (laneN == 1) ? su[k + 1] : 0.f;
        params[256 + t * 2 + 0] = b0;
        params[256 + t * 2 + 1] = b1;
    }
}

// ---------------------------------------------------------------------------
// Kernel I: init per-node segment-max (enc(-inf)) and interleaved denom/numer.
// ---------------------------------------------------------------------------
__global__ void init_kernel(unsigned* __restrict__ segmax, float* __restrict__ dn, int N) {
    int n = blockIdx.x * blockDim.x + threadIdx.x;
    if (n >= N) return;
    segmax[n] = 0x007FFFFFu;  // enc(-INF)
    dn[2 * n]     = 0.f;
    dn[2 * n + 1] = 0.f;
}

// ---------------------------------------------------------------------------
// Kernel A: per-edge dual 16-dot via chained V_WMMA_F32_16X16X4_F32.
// Each wave handles 16 edges: A = edge_attr tile (16x16 fed as 4 K-slabs of
// 16x4), B from the precomputed per-lane image (col0 = v, col1 = u).
// D column 0 -> a_lin[e] (ws), column 1 -> r[e] (stored in d_out, reused).
// ---------------------------------------------------------------------------
__global__ void __launch_bounds__(256)
edge_dot_kernel(const float* __restrict__ ea, const float* __restrict__ params,
                float* __restrict__ a_lin, float* __restrict__ r_out, int E) {
    const int wave = (blockIdx.x * blockDim.x + threadIdx.x) >> 5;
    const int lane = threadIdx.x & 31;
    const long e0 = (long)wave * 16;
    if (e0 >= E) return;  // wave-uniform exit

    if (e0 + 16 <= (long)E) {
        // WMMA path: EXEC is all-ones here (branch above is wave-uniform).
        const int half  = lane >> 4;   // 0: K pair {0,1}; 1: K pair {2,3}
        const int laneN = lane & 15;   // output column / A-row
        const float* __restrict__ arow = ea + (e0 + laneN) * 16 + 2 * half;
        const float* __restrict__ bimg = params + 256 + lane * 2;
        v8f acc = {};
        #pragma unroll
        for (int i = 0; i < 4; ++i) {
            v2f a = *(const v2f*)(arow + 4 * i);
            v2f b = *(const v2f*)(bimg + i * 64);
            acc = __builtin_amdgcn_wmma_f32_16x16x4_f32(false, a, false, b,
                                                        (short)0, acc, false, false);
        }
        // D layout: VGPR j, lanes 0-15 hold D[M=j, N=lane]; lanes 16-31 D[M=j+8].
        if (laneN < 2) {
            float* dst = ((laneN == 0) ? a_lin : r_out) + e0 + 8 * half;
            *(float4*)(dst)     = make_float4(acc[0], acc[1], acc[2], acc[3]);
            *(float4*)(dst + 4) = make_float4(acc[4], acc[5], acc[6], acc[7]);
        }
    } else {
        // Scalar tail (unused for E % 16 == 0, kept for generality).
        long e = e0 + lane;
        if (e < (long)E) {
            const float* __restrict__ v = params + 16;
            const float* __restrict__ u = params + 32;
            const float* row = ea + e * 16;
            float sa = 0.f, sr = 0.f;
            #pragma unroll
            for (int k = 0; k < 16; ++k) { sa += row[k] * v[k]; sr += row[k] * u[k]; }
            a_lin[e] = sa;
            r_out[e] = sr;
        }
    }
}

// ---------------------------------------------------------------------------
// Kernel 1: alpha = leaky_relu(x[src]*cs + x[dst]*cd + a_lin), in-place into
// a_lin buffer; segment-max via uint atomicMax.
// ---------------------------------------------------------------------------
__global__ void alpha_kernel(const int* __restrict__ ei, const float* __restrict__ x,
                             const float* __restrict__ params, float* __restrict__ alpha,
                             unsigned* __restrict__ segmax, int E) {
    int e = blockIdx.x * blockDim.x + threadIdx.x;
    if (e >= E) return;
    const int s = ei[e], d = ei[E + e];
    float a = alpha[e] + x[s] * params[0] + x[d] * params[1];
    a = (a > 0.f) ? a : NEG_SLOPE * a;
    alpha[e] = a;
    atomicMax(&segmax[d], enc_f(a));
}

// ---------------------------------------------------------------------------
// Kernel 2: unnorm = exp(alpha - segmax[dst]); fused accumulation of the
// softmax denominator and the scalar aggregation numerator (interleaved).
// ---------------------------------------------------------------------------
__global__ void accum_kernel(const int* __restrict__ ei, const float* __restrict__ x,
                             const float* __restrict__ alpha, const unsigned* __restrict__ segmax,
                             float* __restrict__ dn, int E) {
    int e = blockIdx.x * blockDim.x + threadIdx.x;
    if (e >= E) return;
    const int s = ei[e], d = ei[E + e];
    const float un = expf(alpha[e] - dec_f(segmax[d]));
    atomicAdd(&dn[2 * d],     un);
    atomicAdd(&dn[2 * d + 1], un * x[s]);
}

// ---------------------------------------------------------------------------
// Kernel 3: per-node s = numer/denom; x2 = relu(s*W + bias);
// p[n] = x2 . m0; q[n] = x2 . m1.
// ---------------------------------------------------------------------------
__global__ void node_kernel(const float* __restrict__ params, const float* __restrict__ dn,
                            float* __restrict__ p, float* __restrict__ q, int N) {
    int n = blockIdx.x * blockDim.x + threadIdx.x;
    if (n >= N) return;
    const float den = dn[2 * n];
    const float s = (den > 0.f) ? dn[2 * n + 1] / den : 0.f;
    const float* __restrict__ W    = params + 64;
    const float* __restrict__ bias = params + 96;
    const float* __restrict__ m0   = params + 128;
    const float* __restrict__ m1   = params + 160;
    float pp = 0.f, qq = 0.f;
    #pragma unroll
    for (int c = 0; c < 32; ++c) {
        const float x2 = fmaxf(fmaf(s, W[c], bias[c]), 0.f);
        pp = fmaf(x2, m0[c], pp);
        qq = fmaf(x2, m1[c], qq);
    }
    p[n] = pp;
    q[n] = qq;
}

// ---------------------------------------------------------------------------
// Kernel 4: out[e] = sigmoid(p[src] + q[dst] + r[e] + mlp_b). r lives in
// d_out (written by edge_dot_kernel); read-then-overwrite same element.
// ---------------------------------------------------------------------------
__global__ void out_kernel(const int* __restrict__ ei, const float* __restrict__ p,
                           const float* __restrict__ q, const float* __restrict__ params,
                           float* __restrict__ out, int E) {
    int e = blockIdx.x * blockDim.x + threadIdx.x;
    if (e >= E) return;
    const int s = ei[e], d = ei[E + e];
    const float z = p[s] + q[d] + out[e] + params[2];
    out[e] = 1.f / (1.f + expf(-z));
}

extern "C" void kernel_launch(void* const* d_in, const int* in_sizes, int n_in,
                              void* d_out, int out_size, void* d_ws, size_t ws_size,
                              hipStream_t stream) {
    const float* x        = (const float*)d_in[0];
    const int*   ei       = (const int*)  d_in[1];   // int32 (JAX x64 disabled)
    const float* ea       = (const float*)d_in[2];
    const float* W        = (const float*)d_in[3];
    const float* att_src  = (const float*)d_in[4];
    const float* att_dst  = (const float*)d_in[5];
    const float* linW     = (const float*)d_in[6];
    const float* attE     = (const float*)d_in[7];
    const float* bias     = (const float*)d_in[8];
    const float* mlpW     = (const float*)d_in[9];
    const float* mlpb     = (const float*)d_in[10];
    float* out = (float*)d_out;

    const int N = in_sizes[0];      // 100000
    const int E = in_sizes[1] / 2;  // 3200000

    // Workspace layout (floats): [0,512) params+bimg | [512,512+E) a_lin/alpha
    // | N uints segmax | 2N floats dn | N floats p | N floats q.
    float*    wsf    = (float*)d_ws;
    float*    params = wsf;
    float*    alpha  = wsf + 512;                 // a_lin, overwritten with alpha
    unsigned* segmax = (unsigned*)(wsf + 512 + (size_t)E);
    float*    dn     = wsf + 512 + (size_t)E + (size_t)N;
    float*    p      = wsf + 512 + (size_t)E + 3 * (size_t)N;
    float*    q      = wsf + 512 + (size_t)E + 4 * (size_t)N;

    const int BT = 256;
    const int gN = (N + BT - 1) / BT;
    const int gE = (E + BT - 1) / BT;
    // edge_dot: 16 edges per wave, 8 waves per 256-thread block -> 128 edges/block
    const int gD = (E + 127) / 128;

    prep_kernel<<<1, 128, 0, stream>>>(W, att_src, att_dst, linW, attE, bias, mlpW, mlpb, params);
    init_kernel<<<gN, BT, 0, stream>>>(segmax, dn, N);
    edge_dot_kernel<<<gD, BT, 0, stream>>>(ea, params, alpha, out, E);
    alpha_kernel<<<gE, BT, 0, stream>>>(ei, x, params, alpha, segmax, E);
    accum_kernel<<<gE, BT, 0, stream>>>(ei, x, alpha, segmax, dn, E);
    node_kernel<<<gN, BT, 0, stream>>>(params, dn, p, q, N);
    out_kernel<<<gE, BT, 0, stream>>>(ei, p, q, params, out, E);
}